// HRWKV_7_24275155157085
// MI455X (gfx1250) — compile-verified
//
#include <hip/hip_runtime.h>
#include <hip/hip_bf16.h>
#include <math.h>
#include <stdint.h>

#define BB 2
#define TT 1024
#define CC 2048
#define HH 32
#define NN 64
#define KVHn 8
#define KVD 512          // KVHn * NN
#define BT (BB*TT)       // 2048 tokens
#define HN (HH*NN)       // 2048
#define LORA_SUM 576
#define RKV_N 3072
#define EPSV 1e-6f

typedef float v2f __attribute__((ext_vector_type(2)));
typedef float v8f __attribute__((ext_vector_type(8)));

__device__ __forceinline__ float sigf(float x) { return 1.0f / (1.0f + expf(-x)); }

// -- CDNA5 async-copy helpers ------------------------------------------------
// LDS offset of a generic pointer to __shared__: ISA 10.2 — for LDS-aperture
// flat addresses, addr[31:0] is the LDS byte offset.
__device__ __forceinline__ uint32_t lds_off32(const void* p) {
  return (uint32_t)(uintptr_t)p;
}
// GLOBAL_LOAD_ASYNC_TO_LDS_B32 (GV mode): each lane copies 4 bytes from the 64-bit global address in vaddr to the LDS address in vdst.
// Tracked by ASYNCcnt; bypasses VGPRs entirely.
__device__ __forceinline__ void async_ld_b32(uint32_t lds_addr, const float* gptr) {
  asm volatile("global_load_async_to_lds_b32 %0, %1, off"
               :: "v"(lds_addr), "v"(gptr) : "memory");
}
__device__ __forceinline__ void wait_async0() {
  asm volatile("s_wait_asynccnt 0x0" ::: "memory");
}

// ---------------------------------------------------------------------------
// Generic fp32 WMMA GEMM: C[M x N] = A[M x K] @ B[K x N]
// One wave -> one 16x16 tile, K-loop in steps of 4 via v_wmma_f32_16x16x4_f32.
// All launch dims divide evenly (M % (16*WAVES) == 0, N % 16 == 0, K % 4 == 0)
// so EXEC stays all-ones across the WMMA (ISA requirement).
// ---------------------------------------------------------------------------
template<int WAVES>
__global__ void gemm_wmma_f32(const float* __restrict__ A, const float* __restrict__ B,
                              float* __restrict__ C, int K, int lda, int ldb, int ldc) {
  const int lane = threadIdx.x & 31;
  const int wave = threadIdx.x >> 5;
  const int l16  = lane & 15;
  const int half = lane >> 4;                 // 0: lanes 0-15, 1: lanes 16-31
  const int tn = blockIdx.x * 16;
  const int tm = (blockIdx.y * WAVES + wave) * 16;

  const float* Ar = A + (size_t)(tm + l16) * lda;  // row m = tm + l16
  const float* Bc = B + tn + l16;                  // col n = tn + l16

  v8f acc = {0.f, 0.f, 0.f, 0.f, 0.f, 0.f, 0.f, 0.f};
  for (int k0 = 0; k0 < K; k0 += 4) {
    if ((k0 & 63) == 0 && k0 + 64 < K) {           // uniform branch; L2-resident
      __builtin_prefetch(Ar + k0 + 64, 0, 3);      // -> global_prefetch_b8
      __builtin_prefetch(Bc + (size_t)(k0 + 64) * ldb, 0, 3);
    }
    const int ka = k0 + half * 2;        // lanes 0-15: k0,k0+1 ; lanes 16-31: k0+2,k0+3
    v2f af; af.x = Ar[ka];               af.y = Ar[ka + 1];
    v2f bf; bf.x = Bc[(size_t)ka * ldb]; bf.y = Bc[(size_t)(ka + 1) * ldb];
    acc = __builtin_amdgcn_wmma_f32_16x16x4_f32(false, af, false, bf,
                                                (short)0, acc, false, false);
  }
  // C/D layout: VGPR g -> (M = half*8 + g, N = l16)
  float* Cp = C + (size_t)(tm + half * 8) * ldc + tn + l16;
#pragma unroll
  for (int g = 0; g < 8; ++g) Cp[(size_t)g * ldc] = acc[g];
}

// ---------------------------------------------------------------------------
// RMSNorm over C=2048 per token
// ---------------------------------------------------------------------------
__global__ void rms_x_kernel(const float* __restrict__ xin, const float* __restrict__ w,
                             float* __restrict__ out) {
  const int t = blockIdx.x;
  const float* xr = xin + (size_t)t * CC;
  float ssq = 0.f;
  for (int i = threadIdx.x; i < CC; i += 256) { float v = xr[i]; ssq += v * v; }
  __shared__ float red[256];
  red[threadIdx.x] = ssq; __syncthreads();
  for (int s = 128; s > 0; s >>= 1) {
    if (threadIdx.x < s) red[threadIdx.x] += red[threadIdx.x + s];
    __syncthreads();
  }
  const float scale = rsqrtf(red[0] / (float)CC + EPSV);
  for (int i = threadIdx.x; i < CC; i += 256) out[(size_t)t * CC + i] = w[i] * xr[i] * scale;
}

// tanh(xw) -> T1 (BT x 96), sigmoid(xg) -> T2 (BT x 256)
__global__ void lora_act_kernel(const float* __restrict__ L,
                                float* __restrict__ T1, float* __restrict__ T2) {
  const int t = blockIdx.x;
  const float* l = L + (size_t)t * LORA_SUM;
  for (int i = threadIdx.x; i < 96;  i += 256) T1[(size_t)t * 96  + i] = tanhf(l[i]);
  for (int i = threadIdx.x; i < 256; i += 256) T2[(size_t)t * 256 + i] = sigf(l[256 + i]);
}

// w' = -softplus(-(wraw+w0)) - 0.5 ; a = sigmoid(araw+a0)
__global__ void wa_post_kernel(float* __restrict__ W, float* __restrict__ A,
                               const float* __restrict__ w0, const float* __restrict__ a0) {
  size_t i = (size_t)blockIdx.x * blockDim.x + threadIdx.x;
  if (i >= (size_t)BT * HN) return;
  const int col = (int)(i % HN);
  const float wr = W[i] + w0[col];
  const float z  = -wr;                                   // softplus(z) stable
  const float sp = fmaxf(z, 0.f) + log1pf(expf(-fabsf(z)));
  W[i] = -sp - 0.5f;
  A[i] = sigf(A[i] + a0[col]);
}

// r: bias + per-head rmsnorm(ln_r) + RoPE. One wave per (token, head);
// lane holds n and n+32 -> rotate_half pairing is intra-thread.
__global__ void rproc_kernel(const float* __restrict__ RKV, const float* __restrict__ Rb,
                             const float* __restrict__ lnr, const float* __restrict__ cosb,
                             const float* __restrict__ sinb, float* __restrict__ Rout) {
  const int blk = blockIdx.x, tok = blk / HH, h = blk % HH, lane = threadIdx.x;
  const int j0 = h * NN + lane, j1 = j0 + 32;
  float x1 = RKV[(size_t)tok * RKV_N + j0] + Rb[j0];
  float x2 = RKV[(size_t)tok * RKV_N + j1] + Rb[j1];
  float ssq = x1 * x1 + x2 * x2;
  for (int off = 16; off > 0; off >>= 1) ssq += __shfl_xor(ssq, off, 32);
  const float scale = rsqrtf(ssq / (float)NN + EPSV);
  const float r1 = lnr[lane] * x1 * scale, r2 = lnr[lane + 32] * x2 * scale;
  const float c1 = cosb[(size_t)tok * NN + lane],      s1 = sinb[(size_t)tok * NN + lane];
  const float c2 = cosb[(size_t)tok * NN + lane + 32], s2 = sinb[(size_t)tok * NN + lane + 32];
  Rout[(size_t)tok * HN + j0] = r1 * c1 - r2 * s1;
  Rout[(size_t)tok * HN + j1] = r2 * c2 + r1 * s2;
}

// k: bias + rmsnorm(ln_k) + RoPE + k_first mixing. One wave per (token, kv-head).
__global__ void kproc_kernel(const float* __restrict__ RKV, const float* __restrict__ Kb,
                             const float* __restrict__ lnk, const float* __restrict__ cosb,
                             const float* __restrict__ sinb, const float* __restrict__ Kmix,
                             const float* __restrict__ k0v, const float* __restrict__ kfirst,
                             float* __restrict__ Kout) {
  const int blk = blockIdx.x, tok = blk / KVHn, g = blk % KVHn, lane = threadIdx.x;
  const int j0 = g * NN + lane, j1 = j0 + 32;
  float x1 = RKV[(size_t)tok * RKV_N + HN + j0] + Kb[j0];
  float x2 = RKV[(size_t)tok * RKV_N + HN + j1] + Kb[j1];
  float ssq = x1 * x1 + x2 * x2;
  for (int off = 16; off > 0; off >>= 1) ssq += __shfl_xor(ssq, off, 32);
  const float scale = rsqrtf(ssq / (float)NN + EPSV);
  const float k1 = lnk[lane] * x1 * scale, k2 = lnk[lane + 32] * x2 * scale;
  const float c1 = cosb[(size_t)tok * NN + lane],      s1 = sinb[(size_t)tok * NN + lane];
  const float c2 = cosb[(size_t)tok * NN + lane + 32], s2 = sinb[(size_t)tok * NN + lane + 32];
  const float kr1 = k1 * c1 - k2 * s1;
  const float kr2 = k2 * c2 + k1 * s2;
  const float sg1 = sigf(Kmix[(size_t)tok * KVD + j0] + k0v[j0]);
  const float sg2 = sigf(Kmix[(size_t)tok * KVD + j1] + k0v[j1]);
  Kout[(size_t)tok * KVD + j0] = kr1 + (kfirst[(size_t)tok * KVD + j0] - kr1) * sg1;
  Kout[(size_t)tok * KVD + j1] = kr2 + (kfirst[(size_t)tok * KVD + j1] - kr2) * sg2;
}

// v: bias + v_first mixing (no norm/rope)
__global__ void vproc_kernel(const float* __restrict__ RKV, const float* __restrict__ Vb,
                             const float* __restrict__ Vmix, const float* __restrict__ v0v,
                             const float* __restrict__ vfirst, float* __restrict__ Vout) {
  size_t i = (size_t)blockIdx.x * blockDim.x + threadIdx.x;
  if (i >= (size_t)BT * KVD) return;
  const int tok = (int)(i / KVD), j = (int)(i % KVD);
  const float v = RKV[(size_t)tok * RKV_N + HN + KVD + j] + Vb[j];
  const float sg = sigf(Vmix[i] + v0v[j]);
  Vout[i] = v + (vfirst[i] - v) * sg;
}

// ---------------------------------------------------------------------------
// RWKV-7 scan: one block per (b,h); 64 threads (2 waves); state row per thread
// in VGPRs. Per-t vectors are DOUBLE-BUFFERED in LDS via async global->LDS
// copies (ASYNCcnt path): while step t computes, step t+1's r/w/a/k/v stream
// into the alternate buffer, hiding load latency on the serial critical path.
// ---------------------------------------------------------------------------
__global__ void rwkv_scan_kernel(const float* __restrict__ R,  const float* __restrict__ Wp,
                                 const float* __restrict__ Ap, const float* __restrict__ Kb,
                                 const float* __restrict__ Vb, const float* __restrict__ rk,
                                 const float* __restrict__ state0,
                                 float* __restrict__ XX, float* __restrict__ stateOut) {
  const int bh = blockIdx.x;
  const int b = bh / HH, h = bh % HH;
  const int g = h / (HH / KVHn);           // REP = 4
  const int tid = threadIdx.x;             // row v = tid
  __shared__ float stage[2][5][NN];        // [buf][r,w,a,k,v][n]
  __shared__ float lkt[NN], lbb[NN], lkk[NN], lwe[NN], lrk[NN];
  lrk[tid] = rk[h * NN + tid];

  float S[NN];
  const float* s0 = state0 + ((size_t)bh * NN + tid) * NN;
#pragma unroll
  for (int j = 0; j < NN; ++j) S[j] = s0[j];

  const size_t hbase = (size_t)b * TT * HN  + (size_t)h * NN + tid;  // step: HN
  const size_t gbase = (size_t)b * TT * KVD + (size_t)g * NN + tid;  // step: KVD

  // stage t = 0
  async_ld_b32(lds_off32(&stage[0][0][tid]), R  + hbase);
  async_ld_b32(lds_off32(&stage[0][1][tid]), Wp + hbase);
  async_ld_b32(lds_off32(&stage[0][2][tid]), Ap + hbase);
  async_ld_b32(lds_off32(&stage[0][3][tid]), Kb + gbase);
  async_ld_b32(lds_off32(&stage[0][4][tid]), Vb + gbase);
  wait_async0();
  __syncthreads();

  for (int t = 0; t < TT; ++t) {
    const int cur = t & 1, nxt = cur ^ 1;
    if (t + 1 < TT) {                      // prefetch next step into other buffer
      const size_t hn = hbase + (size_t)(t + 1) * HN;
      const size_t gn = gbase + (size_t)(t + 1) * KVD;
      async_ld_b32(lds_off32(&stage[nxt][0][tid]), R  + hn);
      async_ld_b32(lds_off32(&stage[nxt][1][tid]), Wp + hn);
      async_ld_b32(lds_off32(&stage[nxt][2][tid]), Ap + hn);
      async_ld_b32(lds_off32(&stage[nxt][3][tid]), Kb + gn);
      async_ld_b32(lds_off32(&stage[nxt][4][tid]), Vb + gn);
    }
    const float* lr = stage[cur][0];
    const float* lw = stage[cur][1];
    const float* la = stage[cur][2];
    const float* lk = stage[cur][3];
    const float* lv = stage[cur][4];

    float ssq = 0.f;
#pragma unroll
    for (int j = 0; j < NN; ++j) ssq += lk[j] * lk[j];
    const float inv = 1.0f / (sqrtf(ssq) + 1e-12f);
    const float kkv = lk[tid] * inv;
    lkk[tid] = kkv;                                   // kk (pre-decay k)
    lkt[tid] = lk[tid] * (1.0f - lw[tid] + la[tid]);  // final k_t
    lbb[tid] = kkv * la[tid];                         // b_t = kk * a
    lwe[tid] = expf(-expf(lw[tid]));                  // exp(wlog), wlog = -exp(w')
    __syncthreads();

    float sa = 0.f;                                   // sa = S . a_t, a_t = -kk
#pragma unroll
    for (int j = 0; j < NN; ++j) sa -= S[j] * lkk[j];
    float bonus = 0.f;                                // (r * k_t * r_k) summed
#pragma unroll
    for (int j = 0; j < NN; ++j) bonus += lr[j] * lkt[j] * lrk[j];

    const float vt = lv[tid];
    float o = 0.f;
#pragma unroll
    for (int j = 0; j < NN; ++j) {
      S[j] = S[j] * lwe[j] + sa * lbb[j] + vt * lkt[j];
      o += S[j] * lr[j];
    }
    XX[hbase + (size_t)t * HN] = o * 0.125f + bonus * vt;   // N^{-1/2} = 1/8

    wait_async0();                        // t+1 staging complete (per wave)
    __syncthreads();                      // visible to both waves; cur buf free
  }
  float* so = stateOut + ((size_t)bh * NN + tid) * NN;
#pragma unroll
  for (int j = 0; j < NN; ++j) so[j] = S[j];
}

__global__ void gate_mul_kernel(const float* __restrict__ XXv, const float* __restrict__ G,
                                float* __restrict__ XG) {
  size_t i = (size_t)blockIdx.x * blockDim.x + threadIdx.x;
  if (i >= (size_t)BT * HN) return;
  XG[i] = XXv[i] * G[i];
}

// x_res = x_in + out_proj ; output = rmsnorm(x_res, ln2)
__global__ void final_kernel(const float* __restrict__ xin, const float* __restrict__ OU,
                             const float* __restrict__ ln2, float* __restrict__ out0,
                             float* __restrict__ out5) {
  const int t = blockIdx.x;
  const float* xr = xin + (size_t)t * CC;
  const float* ou = OU  + (size_t)t * CC;
  float loc[CC / 256];
  float ssq = 0.f;
  for (int i = threadIdx.x, c = 0; i < CC; i += 256, ++c) {
    const float v = xr[i] + ou[i];
    loc[c] = v; ssq += v * v;
  }
  __shared__ float red[256];
  red[threadIdx.x] = ssq; __syncthreads();
  for (int s = 128; s > 0; s >>= 1) {
    if (threadIdx.x < s) red[threadIdx.x] += red[threadIdx.x + s];
    __syncthreads();
  }
  const float scale = rsqrtf(red[0] / (float)CC + EPSV);
  for (int i = threadIdx.x, c = 0; i < CC; i += 256, ++c) {
    out5[(size_t)t * CC + i] = loc[c];
    out0[(size_t)t * CC + i] = ln2[i] * loc[c] * scale;
  }
}

__global__ void xlast_kernel(const float* __restrict__ X, float* __restrict__ out1) {
  const int i = blockIdx.x * blockDim.x + threadIdx.x;
  if (i >= BB * CC) return;
  const int b = i / CC, c = i % CC;
  out1[i] = X[((size_t)b * TT + (TT - 1)) * CC + c];
}

// ---------------------------------------------------------------------------
extern "C" void kernel_launch(void* const* d_in, const int* in_sizes, int n_in,
                              void* d_out, int out_size, void* d_ws, size_t ws_size,
                              hipStream_t stream) {
  const float* x_in    = (const float*)d_in[0];
  const float* v_first = (const float*)d_in[1];
  const float* k_first = (const float*)d_in[2];
  const float* state0  = (const float*)d_in[3];
  const float* cosb    = (const float*)d_in[4];
  const float* sinb    = (const float*)d_in[5];
  const float* ln1     = (const float*)d_in[6];
  const float* ln2     = (const float*)d_in[7];
  const float* ln_r    = (const float*)d_in[8];
  const float* ln_k    = (const float*)d_in[9];
  const float* wavgk1  = (const float*)d_in[10];
  const float* w0      = (const float*)d_in[11];
  const float* w2      = (const float*)d_in[12];
  const float* a0      = (const float*)d_in[13];
  const float* a2      = (const float*)d_in[14];
  const float* v0      = (const float*)d_in[15];
  const float* v2      = (const float*)d_in[16];
  const float* k0      = (const float*)d_in[17];
  const float* k2      = (const float*)d_in[18];
  const float* g2      = (const float*)d_in[19];
  const float* r_k     = (const float*)d_in[20];
  const float* RKV_W   = (const float*)d_in[21];
  const float* O_W     = (const float*)d_in[22];
  const float* R_bias  = (const float*)d_in[23];
  const float* K_bias  = (const float*)d_in[24];
  const float* V_bias  = (const float*)d_in[25];

  float* out = (float*)d_out;            // concatenated tuple, return order:
  float* o_output = out;                                   // (B,T,C)
  float* o_xlast  = o_output + (size_t)BT * CC;            // (B,C)
  float* o_state  = o_xlast  + (size_t)BB * CC;            // (B,H,N,N)
  float* o_vfirst = o_state  + (size_t)BB * HH * NN * NN;  // (B,T,KVH,N)
  float* o_kfirst = o_vfirst + (size_t)BT * KVD;
  float* o_xres   = o_kfirst + (size_t)BT * KVD;           // (B,T,C)

  float* ws = (float*)d_ws;
  float* X    = ws; ws += (size_t)BT * CC;       // rmsnorm(x_in)
  float* L    = ws; ws += (size_t)BT * LORA_SUM; // x @ wavgk1
  float* RKV  = ws; ws += (size_t)BT * RKV_N;    // x @ RKV_W
  float* Wp   = ws; ws += (size_t)BT * HN;       // w' (post-softplus)
  float* Apb  = ws; ws += (size_t)BT * HN;       // a (sigmoid)
  float* Gp   = ws; ws += (size_t)BT * HN;       // sigmoid(xg) @ g2
  float* Rb   = ws; ws += (size_t)BT * HN;       // processed r
  float* Kb   = ws; ws += (size_t)BT * KVD;      // processed k (per kv-head)
  float* Vb   = ws; ws += (size_t)BT * KVD;      // processed v
  float* Vmix = ws; ws += (size_t)BT * KVD;      // xv @ v2 (raw)
  float* Kmix = ws; ws += (size_t)BT * KVD;      // xk @ k2 (raw)
  float* T1   = ws; ws += (size_t)BT * 96;       // tanh(xw)
  float* T2   = ws; ws += (size_t)BT * 256;      // sigmoid(xg)
  float* XXb  = ws; ws += (size_t)BT * HN;       // scan output + bonus
  float* XG   = ws; ws += (size_t)BT * HN;       // gated
  float* OU   = ws; ws += (size_t)BT * CC;       // XG @ O_W

  rms_x_kernel<<<BT, 256, 0, stream>>>(x_in, ln1, X);

  // big GEMMs off X (fp32 WMMA 16x16x4)
  gemm_wmma_f32<4><<<dim3(LORA_SUM/16, BT/64), 128, 0, stream>>>(X, wavgk1, L,   CC, CC, LORA_SUM, LORA_SUM);
  gemm_wmma_f32<4><<<dim3(RKV_N/16,    BT/64), 128, 0, stream>>>(X, RKV_W,  RKV, CC, CC, RKV_N,    RKV_N);

  lora_act_kernel<<<BT, 256, 0, stream>>>(L, T1, T2);

  // LoRA GEMMs (A slices of L use lda=576 strides)
  gemm_wmma_f32<4><<<dim3(HN/16,  BT/64), 128, 0, stream>>>(T1,      w2, Wp,   96,  96,  HN,  HN);
  gemm_wmma_f32<4><<<dim3(HN/16,  BT/64), 128, 0, stream>>>(L + 96,  a2, Apb,  96,  576, HN,  HN);
  gemm_wmma_f32<4><<<dim3(KVD/16, BT/64), 128, 0, stream>>>(L + 192, v2, Vmix, 64,  576, KVD, KVD);
  gemm_wmma_f32<4><<<dim3(KVD/16, BT/64), 128, 0, stream>>>(L + 512, k2, Kmix, 64,  576, KVD, KVD);
  gemm_wmma_f32<4><<<dim3(HN/16,  BT/64), 128, 0, stream>>>(T2,      g2, Gp,   256, 256, HN,  HN);

  wa_post_kernel<<<(BT*HN + 255)/256, 256, 0, stream>>>(Wp, Apb, w0, a0);
  rproc_kernel<<<BT*HH,  32, 0, stream>>>(RKV, R_bias, ln_r, cosb, sinb, Rb);
  kproc_kernel<<<BT*KVHn, 32, 0, stream>>>(RKV, K_bias, ln_k, cosb, sinb, Kmix, k0, k_first, Kb);
  vproc_kernel<<<(BT*KVD + 255)/256, 256, 0, stream>>>(RKV, V_bias, Vmix, v0, v_first, Vb);

  rwkv_scan_kernel<<<BB*HH, NN, 0, stream>>>(Rb, Wp, Apb, Kb, Vb, r_k, state0, XXb, o_state);

  gate_mul_kernel<<<(BT*HN + 255)/256, 256, 0, stream>>>(XXb, Gp, XG);
  gemm_wmma_f32<4><<<dim3(CC/16, BT/64), 128, 0, stream>>>(XG, O_W, OU, HN, HN, CC, CC);

  final_kernel<<<BT, 256, 0, stream>>>(x_in, OU, ln2, o_output, o_xres);
  xlast_kernel<<<(BB*CC + 255)/256, 256, 0, stream>>>(X, o_xlast);

  hipMemcpyAsync(o_vfirst, v_first, sizeof(float)*(size_t)BT*KVD, hipMemcpyDeviceToDevice, stream);
  hipMemcpyAsync(o_kfirst, k_first, sizeof(float)*(size_t)BT*KVD, hipMemcpyDeviceToDevice, stream);
}